// ODConv2D_74199855005816
// MI455X (gfx1250) — compile-verified
//
#include <hip/hip_runtime.h>
#include <math.h>

// ---------------------------------------------------------------------------
// ODConv2D fused implementation for gfx1250 (MI455X)
// B=16, C=COUT=256, H=W=64, K=3, NC=6, stride=1, pad=1
// ---------------------------------------------------------------------------

#define Bn    16
#define Cn    256
#define COUTn 256
#define Hn    64
#define Wn    64
#define Kn    3
#define KKn   9
#define NCn   6
#define HWn   (Hn * Wn)           // 4096

typedef __attribute__((ext_vector_type(2))) float v2f;
typedef __attribute__((ext_vector_type(8))) float v8f;

// --------------------------- K1: global average pool -----------------------
__global__ __launch_bounds__(256)
void gap_kernel(const float* __restrict__ x, float* __restrict__ ctx) {
    const int bc = blockIdx.x;                    // 0 .. B*C-1
    const float* p = x + ((size_t)bc << 12);      // * 4096
    float s = 0.f;
    for (int i = threadIdx.x; i < HWn; i += 256) s += p[i];
    __shared__ float red[256];
    red[threadIdx.x] = s;
    __syncthreads();
    for (int off = 128; off > 0; off >>= 1) {
        if (threadIdx.x < off) red[threadIdx.x] += red[threadIdx.x + off];
        __syncthreads();
    }
    if (threadIdx.x == 0) ctx[bc] = red[0] * (1.0f / (float)HWn);
}

// ----------------- K2: sigmoid attention heads (sp/in/k/o) -----------------
// per-batch outputs: 9 (sp) + 256 (in) + 2304 (k) + 256 (o) = 2825
#define PERB 2825
__global__ __launch_bounds__(256)
void attn_sigmoid_kernel(const float* __restrict__ ctx,
                         const float* __restrict__ w_sp, const float* __restrict__ b_sp,
                         const float* __restrict__ w_in, const float* __restrict__ b_in,
                         const float* __restrict__ w_k,  const float* __restrict__ b_k,
                         const float* __restrict__ w_o,  const float* __restrict__ b_o,
                         float* __restrict__ a_sp, float* __restrict__ a_in,
                         float* __restrict__ a_k,  float* __restrict__ a_o) {
    const int tid = blockIdx.x * 256 + threadIdx.x;
    if (tid >= Bn * PERB) return;
    const int b = tid / PERB;
    const int r = tid % PERB;
    const float* w;  float bias;  float* outp;
    if (r < 9)            { int i = r;        w = w_sp + i * Cn; bias = b_sp[i]; outp = a_sp + b * KKn       + i; }
    else if (r < 265)     { int i = r - 9;    w = w_in + i * Cn; bias = b_in[i]; outp = a_in + b * Cn        + i; }
    else if (r < 2569)    { int i = r - 265;  w = w_k  + i * Cn; bias = b_k[i];  outp = a_k  + b * (Cn*KKn)  + i; }
    else                  { int i = r - 2569; w = w_o  + i * Cn; bias = b_o[i];  outp = a_o  + b * COUTn     + i; }
    const float* c = ctx + b * Cn;
    float dot = bias;
    #pragma unroll 8
    for (int i = 0; i < Cn; ++i) dot += c[i] * w[i];
    *outp = 1.0f / (1.0f + __expf(-dot));
}

// ------------------------- K3: softmax channel head ------------------------
__global__ void attn_softmax_kernel(const float* __restrict__ ctx,
                                    const float* __restrict__ w_c,
                                    const float* __restrict__ b_c,
                                    float* __restrict__ a_c) {
    const int b = threadIdx.x;
    if (b >= Bn) return;
    const float* c = ctx + b * Cn;
    float lg[NCn];
    float mx = -3.0e38f;
    for (int n = 0; n < NCn; ++n) {
        float d = b_c[n];
        const float* w = w_c + n * Cn;
        for (int i = 0; i < Cn; ++i) d += c[i] * w[i];
        lg[n] = d;
        mx = fmaxf(mx, d);
    }
    float sum = 0.f;
    for (int n = 0; n < NCn; ++n) { lg[n] = __expf(lg[n] - mx); sum += lg[n]; }
    const float inv = 1.0f / sum;
    for (int n = 0; n < NCn; ++n) a_c[b * NCn + n] = lg[n] * inv;
}

// ---------------- K4: build per-sample depthwise kernel dyn ----------------
__global__ __launch_bounds__(256)
void dyn_kernel(const float* __restrict__ bank, const float* __restrict__ a_c,
                const float* __restrict__ a_sp, const float* __restrict__ a_in,
                const float* __restrict__ a_k,  float* __restrict__ dyn) {
    const int tid = blockIdx.x * 256 + threadIdx.x;
    if (tid >= Bn * Cn * KKn) return;
    const int b   = tid / (Cn * KKn);
    const int rem = tid % (Cn * KKn);   // c*9 + kk
    const int c   = rem / KKn;
    const int kk  = rem % KKn;
    float agg = 0.f;
    #pragma unroll
    for (int n = 0; n < NCn; ++n)
        agg += bank[n * (Cn * KKn) + rem] * a_c[b * NCn + n];
    dyn[tid] = agg * a_sp[b * KKn + kk] * a_in[b * Cn + c] * a_k[b * (Cn * KKn) + rem];
}

// -------- K5: fused depthwise-3x3 + pointwise GEMM via f32 WMMA ------------
// One block per (b, h): computes out[b, 0..255, h, 0..63].
// 256 threads = 8 waves. Wave w owns M-tiles {2w, 2w+1} x N-tiles {0..3}.
// K loop over C in chunks of 16; dw chunk staged in LDS.
#define LDS_STRIDE 80   // 16 floats past 64: lane-halves (2 rows apart) hit disjoint banks

__global__ __launch_bounds__(256)
void odconv_fused_kernel(const float* __restrict__ x,
                         const float* __restrict__ dyn,
                         const float* __restrict__ pw_w,
                         const float* __restrict__ pw_b,
                         const float* __restrict__ a_o,
                         float* __restrict__ out) {
    const int h    = blockIdx.x;        // 0..63
    const int b    = blockIdx.y;        // 0..15
    const int t    = threadIdx.x;
    const int lane = t & 31;
    const int wave = t >> 5;            // 0..7
    const int lhalf = lane >> 4;        // 0|1 : lane half selects K pair / M+8
    const int l16   = lane & 15;

    __shared__ float lds_dw[16][LDS_STRIDE];

    v8f acc[2][4];
    #pragma unroll
    for (int mi = 0; mi < 2; ++mi)
        #pragma unroll
        for (int nt = 0; nt < 4; ++nt)
            acc[mi][nt] = (v8f)0.0f;

    // depthwise producer mapping: thread -> (channel-in-chunk, 4 pixels)
    const int cl = t >> 4;              // 0..15
    const int pb = (t & 15) << 2;       // 0,4,...,60

    for (int cb = 0; cb < Cn; cb += 16) {
        __syncthreads();                // protect LDS from previous consumers

        // ---- produce dw[cb+cl][pb..pb+3] ----
        {
            const int c = cb + cl;
            const float* xc = x + (((size_t)(b * Cn + c)) << 12);  // x[b,c,0,0]
            const float* d9 = dyn + (b * Cn + c) * KKn;
            float kv[KKn];
            #pragma unroll
            for (int j = 0; j < KKn; ++j) kv[j] = d9[j];
            #pragma unroll
            for (int pi = 0; pi < 4; ++pi) {
                const int p = pb + pi;
                float s = 0.f;
                #pragma unroll
                for (int dy = -1; dy <= 1; ++dy) {
                    const int hh = h + dy;
                    if ((unsigned)hh < (unsigned)Hn) {
                        const float* row = xc + hh * Wn;
                        #pragma unroll
                        for (int dx = -1; dx <= 1; ++dx) {
                            const int ww = p + dx;
                            if ((unsigned)ww < (unsigned)Wn)
                                s += row[ww] * kv[(dy + 1) * 3 + (dx + 1)];
                        }
                    }
                }
                lds_dw[cl][p] = s;
            }
        }
        __syncthreads();

        // ---- consume: 4 WMMA K-steps of 4 over this 16-channel chunk ----
        #pragma unroll
        for (int ks = 0; ks < 4; ++ks) {
            const int kb = ks * 4;            // K offset inside chunk
            const int kg = cb + kb;           // global K (channel)

            // A fragments (pw_w, row-major [COUT][C]):
            // lanes 0-15 -> {K+0,K+1}, lanes 16-31 -> {K+2,K+3}  (16x4 f32 layout)
            v2f afrag[2];
            #pragma unroll
            for (int mi = 0; mi < 2; ++mi) {
                const int row = (wave * 2 + mi) * 16 + l16;
                const float* ap = pw_w + row * Cn + kg + lhalf * 2;
                afrag[mi].x = ap[0];
                afrag[mi].y = ap[1];
            }

            #pragma unroll
            for (int nt = 0; nt < 4; ++nt) {
                // B fragment (4x16): VGPR v, lanes 0-15 -> K=v, lanes 16-31 -> K=v+2
                const int col = nt * 16 + l16;
                v2f bfrag;
                bfrag.x = lds_dw[kb + lhalf * 2 + 0][col];
                bfrag.y = lds_dw[kb + lhalf * 2 + 1][col];
                #pragma unroll
                for (int mi = 0; mi < 2; ++mi) {
                    acc[mi][nt] = __builtin_amdgcn_wmma_f32_16x16x4_f32(
                        /*neg_a=*/false, afrag[mi],
                        /*neg_b=*/false, bfrag,
                        /*c_mod=*/(short)0, acc[mi][nt],
                        /*reuse_a=*/false, /*reuse_b=*/false);
                }
            }
        }
    }

    // ---- epilogue: + pw_b, * a_o, store ----
    // C/D layout: VGPR v -> row v (lanes 0-15) / row v+8 (lanes 16-31)
    #pragma unroll
    for (int mi = 0; mi < 2; ++mi) {
        #pragma unroll
        for (int nt = 0; nt < 4; ++nt) {
            #pragma unroll
            for (int v = 0; v < 8; ++v) {
                const int o = (wave * 2 + mi) * 16 + v + lhalf * 8;
                const int p = nt * 16 + l16;
                const float val = (acc[mi][nt][v] + pw_b[o]) * a_o[b * COUTn + o];
                out[(((size_t)(b * COUTn + o)) << 12) + h * Wn + p] = val;
            }
        }
    }
}

// ---------------------------------------------------------------------------
extern "C" void kernel_launch(void* const* d_in, const int* in_sizes, int n_in,
                              void* d_out, int out_size, void* d_ws, size_t ws_size,
                              hipStream_t stream) {
    const float* x    = (const float*)d_in[0];
    const float* bank = (const float*)d_in[1];
    const float* w_sp = (const float*)d_in[2];
    const float* b_sp = (const float*)d_in[3];
    const float* w_in = (const float*)d_in[4];
    const float* b_in = (const float*)d_in[5];
    const float* w_k  = (const float*)d_in[6];
    const float* b_k  = (const float*)d_in[7];
    const float* w_c  = (const float*)d_in[8];
    const float* b_c  = (const float*)d_in[9];
    const float* w_o  = (const float*)d_in[10];
    const float* b_o  = (const float*)d_in[11];
    const float* pw_w = (const float*)d_in[12];
    const float* pw_b = (const float*)d_in[13];
    float* out = (float*)d_out;

    // workspace layout (floats)
    float* ws    = (float*)d_ws;
    float* ctx   = ws;                         // 16*256   = 4096
    float* a_sp  = ctx  + Bn * Cn;             // 16*9     = 144
    float* a_in  = a_sp + Bn * KKn;            // 16*256   = 4096
    float* a_k   = a_in + Bn * Cn;             // 16*2304  = 36864
    float* a_o   = a_k  + Bn * Cn * KKn;       // 16*256   = 4096
    float* a_c   = a_o  + Bn * COUTn;          // 16*6     = 96
    float* dyn   = a_c  + Bn * NCn;            // 16*2304  = 36864

    gap_kernel<<<Bn * Cn, 256, 0, stream>>>(x, ctx);

    const int n_attn = Bn * PERB;
    attn_sigmoid_kernel<<<(n_attn + 255) / 256, 256, 0, stream>>>(
        ctx, w_sp, b_sp, w_in, b_in, w_k, b_k, w_o, b_o,
        a_sp, a_in, a_k, a_o);

    attn_softmax_kernel<<<1, 32, 0, stream>>>(ctx, w_c, b_c, a_c);

    const int n_dyn = Bn * Cn * KKn;
    dyn_kernel<<<(n_dyn + 255) / 256, 256, 0, stream>>>(
        bank, a_c, a_sp, a_in, a_k, dyn);

    dim3 grid(Hn, Bn);
    odconv_fused_kernel<<<grid, 256, 0, stream>>>(x, dyn, pw_w, pw_b, a_o, out);
}